// ImprovedPositionAwareLoss_49735721288422
// MI455X (gfx1250) — compile-verified
//
#include <hip/hip_runtime.h>
#include <hip/hip_bf16.h>

// Streaming weighted-CE + positional pair loss, single-scalar output.
// HBM-bound: ~256 MB read once -> ~11 us floor at 23.3 TB/s on MI455X.
// Chunked 4 elements/thread: b128 NT loads, 1 exp + 1 log1p + 1 rcp per element.

typedef float v2f __attribute__((ext_vector_type(2)));
typedef float v4f __attribute__((ext_vector_type(4)));
typedef int   v4i __attribute__((ext_vector_type(4)));

#define THREADS 256
#define NBLOCKS 4096
#define CHUNK 4

// e = exp(-|d|), shared by nll and sigmoid.
__device__ __forceinline__ float sig_from(float d, float e) {
    float r = __builtin_amdgcn_rcpf(1.0f + e);
    return (d >= 0.0f) ? r : e * r;
}

__global__ __launch_bounds__(THREADS)
void ImprovedPositionAwareLoss_reduce(const float* __restrict__ logits,
                                      const int*   __restrict__ labels,
                                      const int*   __restrict__ batch,
                                      double*      __restrict__ acc,
                                      int N) {
    float a_wnll = 0.0f;   // sum w[y] * nll
    int   a_cnt  = 0;      // count of label==1 (w-sum = N + 6*cnt, exact)
    float a_pos  = 0.0f;   // sum |p1[i+1]-p1[i]| * mask

    const v4f* lg4  = (const v4f*)logits;
    const v4i* lab4 = (const v4i*)labels;
    const v4i* bat4 = (const v4i*)batch;
    const v2f* lg2  = (const v2f*)logits;

    const int nChunks = N >> 2;                  // full 4-element chunks
    const int cstride = THREADS * NBLOCKS;

    for (int c = blockIdx.x * THREADS + threadIdx.x; c < nChunks; c += cstride) {
        // Prefetch next grid-stride iteration's logits (speculative).
        if (c + cstride < nChunks)
            __builtin_prefetch(&logits[8 * (c + cstride)], 0, 0);

        // Single-use stream: non-temporal b128 loads, don't churn L2.
        v4f lgA = __builtin_nontemporal_load(&lg4[2 * c]);       // elems 4c, 4c+1
        v4f lgB = __builtin_nontemporal_load(&lg4[2 * c + 1]);   // elems 4c+2, 4c+3
        v4i lab = __builtin_nontemporal_load(&lab4[c]);
        v4i bat = __builtin_nontemporal_load(&bat4[c]);

        const int base  = c << 2;
        const int ipn   = base + CHUNK;                          // boundary neighbor
        const int valid = (ipn < N) ? 1 : 0;
        const int ipc   = valid ? ipn : (N - 1);
        v2f lgN  = __builtin_nontemporal_load(&lg2[ipc]);
        int labN = __builtin_nontemporal_load(&labels[ipc]);
        int batN = __builtin_nontemporal_load(&batch[ipc]);

        float d[5]; int y[5]; int b[5];
        d[0] = lgA.y - lgA.x;  d[1] = lgA.w - lgA.z;
        d[2] = lgB.y - lgB.x;  d[3] = lgB.w - lgB.z;
        d[4] = lgN.y - lgN.x;
        y[0] = lab.x; y[1] = lab.y; y[2] = lab.z; y[3] = lab.w; y[4] = labN;
        b[0] = bat.x; b[1] = bat.y; b[2] = bat.z; b[3] = bat.w; b[4] = batN;

        float p1[5];
        #pragma unroll
        for (int k = 0; k < 5; ++k) {
            float e = expf(-fabsf(d[k]));                        // shared exp
            p1[k] = sig_from(d[k], e);
            if (k < 4) {
                float t   = log1pf(e);
                float nll = (y[k] ? fmaxf(-d[k], 0.0f) : fmaxf(d[k], 0.0f)) + t;
                a_wnll += (y[k] ? 7.0f : 1.0f) * nll;
                a_cnt  += y[k];
            }
        }
        #pragma unroll
        for (int k = 0; k < 3; ++k) {
            int m = (b[k] == b[k + 1] ? 1 : 0) & (((y[k] | y[k + 1]) != 0) ? 1 : 0);
            a_pos += (float)m * fabsf(p1[k + 1] - p1[k]);
        }
        int m3 = valid & (b[3] == b[4] ? 1 : 0) & (((y[3] | y[4]) != 0) ? 1 : 0);
        a_pos += (float)m3 * fabsf(p1[4] - p1[3]);
    }

    // Tail elements [nChunks*4, N): scalar cleanup by one thread (<= 3 elems).
    if (blockIdx.x == 0 && threadIdx.x == 0) {
        for (int i = nChunks << 2; i < N; ++i) {
            v2f lg = lg2[i];
            float dd = lg.y - lg.x;
            int   yy = labels[i];
            float e  = expf(-fabsf(dd));
            float nll = (yy ? fmaxf(-dd, 0.0f) : fmaxf(dd, 0.0f)) + log1pf(e);
            a_wnll += (yy ? 7.0f : 1.0f) * nll;
            a_cnt  += yy;
            if (i + 1 < N) {
                v2f lgn = lg2[i + 1];
                float dn = lgn.y - lgn.x;
                float en = expf(-fabsf(dn));
                int m = (batch[i] == batch[i + 1] ? 1 : 0) &
                        (((yy | labels[i + 1]) != 0) ? 1 : 0);
                a_pos += (float)m * fabsf(sig_from(dn, en) - sig_from(dd, e));
            }
        }
    }

    // wave32 butterfly reduction
    for (int off = 16; off > 0; off >>= 1) {
        a_wnll += __shfl_down(a_wnll, off, 32);
        a_cnt  += __shfl_down(a_cnt,  off, 32);
        a_pos  += __shfl_down(a_pos,  off, 32);
    }

    __shared__ float s0[THREADS / 32];
    __shared__ int   s1[THREADS / 32];
    __shared__ float s2[THREADS / 32];
    const int wave = threadIdx.x >> 5;
    const int lane = threadIdx.x & 31;
    if (lane == 0) { s0[wave] = a_wnll; s1[wave] = a_cnt; s2[wave] = a_pos; }
    __syncthreads();

    if (threadIdx.x == 0) {
        double t0 = 0.0, t2 = 0.0; long long t1 = 0;
        #pragma unroll
        for (int k = 0; k < THREADS / 32; ++k) { t0 += s0[k]; t1 += s1[k]; t2 += s2[k]; }
        // One f64 atomic per block per accumulator (global_atomic_add_f64).
        atomicAdd(&acc[0], t0);
        atomicAdd(&acc[1], (double)t1);
        atomicAdd(&acc[2], t2);
    }
}

__global__ void ImprovedPositionAwareLoss_finalize(const double* __restrict__ acc,
                                                   float* __restrict__ out,
                                                   int N) {
    if (threadIdx.x == 0 && blockIdx.x == 0) {
        double w_sum = (double)N + 6.0 * acc[1];   // exact: 1 + 6*[y==1]
        double ce    = acc[0] / w_sum;
        out[0] = (float)(ce + 0.1 * acc[2]);
    }
}

extern "C" void kernel_launch(void* const* d_in, const int* in_sizes, int n_in,
                              void* d_out, int out_size, void* d_ws, size_t ws_size,
                              hipStream_t stream) {
    const float* logits = (const float*)d_in[0];   // [N, 2] f32
    const int*   labels = (const int*)d_in[1];     // [N]
    const int*   batch  = (const int*)d_in[2];     // [N]
    const int N = in_sizes[1];                     // labels is flat [N]

    double* acc = (double*)d_ws;                   // [3]: wnll, pos_count, pos

    // Capture-safe zeroing of the accumulators every launch (deterministic).
    hipMemsetAsync(acc, 0, 3 * sizeof(double), stream);

    ImprovedPositionAwareLoss_reduce<<<NBLOCKS, THREADS, 0, stream>>>(
        logits, labels, batch, acc, N);
    ImprovedPositionAwareLoss_finalize<<<1, 32, 0, stream>>>(acc, (float*)d_out, N);
}